// FilterDetection_65085934403666
// MI455X (gfx1250) — compile-verified
//
#include <hip/hip_runtime.h>
#include <stdint.h>

// ---------------------------------------------------------------------------
// FilterDetection (tensorrt path): scores = logits*confs ; boxes = delta2bbox
// Memory-bound: ~398 MB traffic -> ~17us at 23.3 TB/s. Strategy: b128
// streaming everywhere + TDM (tensor_load_to_lds) staging of the anchor tile
// into LDS (anchors reused B=4x per tile).
// Toolchain note: this lane's clang exposes the 6-arg TDM builtin
// (uint32x4, int32x8, int32x4, int32x4, int32x8, i32 cpol).
// ---------------------------------------------------------------------------

typedef __attribute__((ext_vector_type(4))) unsigned int u32x4;
typedef __attribute__((ext_vector_type(8))) int          i32x8;
typedef __attribute__((ext_vector_type(4))) int          i32x4;

#define TILE_N 256

// ---- Kernel 1: scores = logits * confs (vectorized float4 stream) ----------
__global__ __launch_bounds__(256) void scores_mul_v4(
    const float4* __restrict__ logits,
    const float4* __restrict__ confs,
    float4* __restrict__ out, int n4) {
  int i = blockIdx.x * blockDim.x + threadIdx.x;
  if (i < n4) {
    float4 a = logits[i];
    float4 b = confs[i];
    float4 r;
    r.x = a.x * b.x;
    r.y = a.y * b.y;
    r.z = a.z * b.z;
    r.w = a.w * b.w;
    out[i] = r;
  }
}

// Scalar tail (only launched if total % 4 != 0; not the case for B*N*C=32M).
__global__ void scores_mul_tail(const float* __restrict__ logits,
                                const float* __restrict__ confs,
                                float* __restrict__ out, int start, int total) {
  int i = start + blockIdx.x * blockDim.x + threadIdx.x;
  if (i < total) out[i] = logits[i] * confs[i];
}

// ---- Kernel 2: box decode with TDM-staged anchors --------------------------
__global__ __launch_bounds__(TILE_N) void boxes_decode(
    const float4* __restrict__ regress,   // [B*N] float4 (dx,dy,dw,dh)
    const float*  __restrict__ anchors,   // [N*4] floats (x1,y1,x2,y2)
    float4* __restrict__ out,             // [B*N] float4
    int N, int B) {
  __shared__ float4 s_anchors[TILE_N];

  const int n0      = blockIdx.x * TILE_N;
  const int count_n = min(TILE_N, N - n0);

  // Wave 0 issues one Tensor Data Mover load of this block's anchor tile
  // (count_n * 4 floats) from global memory into LDS, then waits on
  // TENSORcnt. Descriptor layout per CDNA5 ISA:
  //   group0: count=1 | lds_addr | global_addr[56:0] | type=2
  //   group1: data_size=4B, tensor_dim0=tile_dim0=nelem (1-D tile),
  //           tensor_dim1=1, dim0_stride=nelem
  if (threadIdx.x < 32) {
    const uint64_t gaddr = (uint64_t)(uintptr_t)(anchors + (size_t)n0 * 4);
    const uint32_t lds   = (uint32_t)(uintptr_t)(&s_anchors[0]);
    const uint32_t nelem = (uint32_t)count_n * 4u;   // float elements

    u32x4 g0;
    g0[0] = 1u;                                            // count=1, user D#
    g0[1] = lds;                                           // lds_addr (bytes)
    g0[2] = (uint32_t)(gaddr & 0xFFFFFFFFull);             // global_addr lo
    g0[3] = (uint32_t)((gaddr >> 32) & 0x01FFFFFFull)      // global_addr hi
            | 0x80000000u;                                 // type = 2 (image)

    i32x8 g1;
    g1[0] = 0x00020000;                                    // data_size = 4B
    g1[1] = (int)((nelem & 0xFFFFu) << 16);                // tensor_dim0 lo16
    g1[2] = (int)(((nelem >> 16) & 0xFFFFu) | (1u << 16)); // dim0 hi | dim1=1
    g1[3] = (int)(nelem << 16);                            // tile_dim0
    g1[4] = 0;                                             // tile_dim1/2 = 0
    g1[5] = (int)nelem;                                    // dim0_stride lo32
    g1[6] = 0;
    g1[7] = 0;

    i32x4 g2 = {0, 0, 0, 0};
    i32x4 g3 = {0, 0, 0, 0};
    i32x8 g4 = {0, 0, 0, 0, 0, 0, 0, 0};

    __builtin_amdgcn_tensor_load_to_lds(g0, g1, g2, g3, g4, /*cpol=*/0);
    __builtin_amdgcn_s_wait_tensorcnt(0);
  }
  __syncthreads();

  const int t = threadIdx.x;
  if (t < count_n) {
    const float MR = 4.135166556742356f;  // |ln(16/1000)|
    const float4 a = s_anchors[t];
    const float px = (a.x + a.z) * 0.5f;
    const float py = (a.y + a.w) * 0.5f;
    const float pw = a.z - a.x;
    const float ph = a.w - a.y;

    const int n = n0 + t;
    for (int b = 0; b < B; ++b) {
      const size_t idx = (size_t)b * (size_t)N + (size_t)n;
      const float4 d = regress[idx];
      const float dw = fminf(fmaxf(d.z, -MR), MR);
      const float dh = fminf(fmaxf(d.w, -MR), MR);
      const float gx = fmaf(pw, d.x, px);
      const float gy = fmaf(ph, d.y, py);
      const float gw = pw * __expf(dw);
      const float gh = ph * __expf(dh);
      float4 o;
      o.x = __saturatef(fmaf(gw, -0.5f, gx));
      o.y = __saturatef(fmaf(gh, -0.5f, gy));
      o.z = __saturatef(fmaf(gw,  0.5f, gx));
      o.w = __saturatef(fmaf(gh,  0.5f, gy));
      out[idx] = o;
    }
  }
}

// ---------------------------------------------------------------------------
extern "C" void kernel_launch(void* const* d_in, const int* in_sizes, int n_in,
                              void* d_out, int out_size, void* d_ws, size_t ws_size,
                              hipStream_t stream) {
  // setup_inputs() order: logits [B,N,C], regress [B,N,4], anchors [N,4],
  // confs [B,N,C]. Output: scores flat then boxes flat.
  const float* logits  = (const float*)d_in[0];
  const float* regress = (const float*)d_in[1];
  const float* anchors = (const float*)d_in[2];
  const float* confs   = (const float*)d_in[3];
  float* out = (float*)d_out;

  const int scores_total = in_sizes[0];        // B*N*C
  const int N  = in_sizes[2] / 4;
  const int BN = in_sizes[1] / 4;
  const int B  = N > 0 ? BN / N : 0;

  float* scores_out = out;
  float* boxes_out  = out + (size_t)scores_total;

  // --- scores: vectorized elementwise multiply ---
  const int n4 = scores_total / 4;
  if (n4 > 0) {
    scores_mul_v4<<<(n4 + 255) / 256, 256, 0, stream>>>(
        (const float4*)logits, (const float4*)confs, (float4*)scores_out, n4);
  }
  const int rem = scores_total - n4 * 4;
  if (rem > 0) {
    scores_mul_tail<<<1, 64, 0, stream>>>(logits, confs, scores_out,
                                          n4 * 4, scores_total);
  }

  // --- boxes: TDM-staged anchor decode ---
  if (N > 0 && B > 0) {
    const int nblk = (N + TILE_N - 1) / TILE_N;
    boxes_decode<<<nblk, TILE_N, 0, stream>>>(
        (const float4*)regress, anchors, (float4*)boxes_out, N, B);
  }
}